// pfc_hpc_WCST_model_8615704396357
// MI455X (gfx1250) — compile-verified
//
#include <hip/hip_runtime.h>
#include <hip/hip_bf16.h>

typedef float v2f __attribute__((ext_vector_type(2)));
typedef float v8f __attribute__((ext_vector_type(8)));

#define NBLK 32     // one block per batch element
#define NTHR 256    // 8 waves of 32
#define H    512
#define NINP 17
#define NTRI 3
#define TSTEPS 24

// ---------------- workspace layout (float indices) ----------------
#define WS_A    ((size_t)0)                 // [32][512][512]
#define WS_RH   ((size_t)8388608)           // [32][512]
#define WS_RP   (WS_RH + NBLK*H)
#define WS_GHPC (WS_RP + NBLK*H)            // rp @ W_h2h_hpc^T
#define WS_GRP  (WS_GHPC + NBLK*H)          // rp @ W_hh_pfc^T
#define WS_GRH  (WS_GRP + NBLK*H)           // rh @ W_h2h_pfc^T
#define WS_GE   (WS_GRH + NBLK*H)           // rp_new @ W_e^T
#define WS_BAR  (WS_GE + NBLK*H)            // barrier counters (as unsigned)

// ---------------- grid barrier (persistent kernel, 32 co-resident blocks) ----
__device__ __forceinline__ void gbar(unsigned* cnt, unsigned* gen) {
  __syncthreads();
  __threadfence();
  if (threadIdx.x == 0) {
    volatile unsigned* vgen = (volatile unsigned*)gen;
    unsigned g = *vgen;
    if (atomicAdd(cnt, 1u) == NBLK - 1u) {
      atomicExch(cnt, 0u);
      __threadfence();
      atomicAdd(gen, 1u);
    } else {
      while (*vgen == g) { __builtin_amdgcn_s_sleep(1); }
    }
  }
  __syncthreads();
  __threadfence();
}

// ---------------- block-wide sum (each thread contributes one partial) ------
__device__ __forceinline__ float blk_sum(float x, float* red) {
#pragma unroll
  for (int o = 16; o; o >>= 1) x += __shfl_xor(x, o, 32);
  __syncthreads();
  if ((threadIdx.x & 31) == 0) red[threadIdx.x >> 5] = x;
  __syncthreads();
  float s = 0.f;
#pragma unroll
  for (int w = 0; w < (NTHR / 32); ++w) s += red[w];
  return s;
}

// ---------------- cooperative WMMA GEMM: Y[0:32][nbase:nbase+16] = S @ W^T --
// S: [32][512] row-major (global), W: [512][512] row-major, Y: [32][512].
// 8 waves: wave = kchunk*2 + mtile; each wave: 16x16 tile, K-chunk of 128.
__device__ __forceinline__ void coop_gemm(const float* __restrict__ S,
                                          const float* __restrict__ W,
                                          float* __restrict__ Yo,
                                          int nbase, float* buf /* 8*256 LDS */) {
  const int lane  = threadIdx.x & 31;
  const int wave  = threadIdx.x >> 5;
  const int mtile = wave & 1;
  const int kch   = wave >> 1;               // 0..3
  const int rowm  = mtile * 16 + (lane & 15);
  const int rown  = nbase + (lane & 15);
  const int csel  = (lane >> 4) * 2;         // K sub-pair select per half-wave
  v8f acc = {};
  const int k0b = kch * 128;
#pragma unroll 4
  for (int k0 = k0b; k0 < k0b + 128; k0 += 4) {
    v2f a = *(const v2f*)(S + rowm * H + k0 + csel);
    v2f bb = *(const v2f*)(W + rown * H + k0 + csel);
    acc = __builtin_amdgcn_wmma_f32_16x16x4_f32(
        /*neg_a=*/false, a, /*neg_b=*/false, bb,
        /*c_mod=*/(short)0, acc, /*reuse_a=*/false, /*reuse_b=*/false);
  }
  // stash partial 16x16 tile: C vgpr r -> row (r + 8*(lane>=16)), col lane%16
  float* my = buf + wave * 256;
#pragma unroll
  for (int r = 0; r < 8; ++r)
    my[((r + (lane >> 4) * 8) << 4) + (lane & 15)] = acc[r];
  __syncthreads();
  // reduce 4 K-chunk partials per M-tile; 512 outputs over 256 threads
  for (int tt = threadIdx.x; tt < 512; tt += NTHR) {
    int mt = tt >> 8, idx = tt & 255;
    float s = buf[(0 * 2 + mt) * 256 + idx] + buf[(1 * 2 + mt) * 256 + idx] +
              buf[(2 * 2 + mt) * 256 + idx] + buf[(3 * 2 + mt) * 256 + idx];
    Yo[((mt << 4) + (idx >> 4)) * H + nbase + (idx & 15)] = s;
  }
  __syncthreads();
}

__global__ void wcst_init_bar(unsigned* bar) { bar[0] = 0u; bar[1] = 0u; }

__global__ __launch_bounds__(NTHR)
void wcst_persistent(const float* __restrict__ Xin, const float* __restrict__ Yin,
                     const float* __restrict__ A_init, const float* __restrict__ rh_init,
                     const float* __restrict__ W_ih_pfc, const float* __restrict__ W_hh_pfc,
                     const float* __restrict__ b_pfc, const float* __restrict__ W_h2h_pfc,
                     const float* __restrict__ W_ih_hpc, const float* __restrict__ W_h2h_hpc,
                     const float* __restrict__ lparam, const float* __restrict__ W_e,
                     const float* __restrict__ b_e, const float* __restrict__ W_pe,
                     const float* __restrict__ b_pe, const float* __restrict__ W_o,
                     const float* __restrict__ b_o, float* __restrict__ out,
                     float* __restrict__ ws) {
  __shared__ float s_rh[H], s_rp[H], s_rhn[H], s_rpn[H];
  __shared__ float s_dotA[H], s_rowsum[H], s_coef[H], s_pre[H];
  __shared__ float s_gemm[8 * 256];
  __shared__ float s_red[NTHR / 32];
  __shared__ float s_inp[NINP];
  __shared__ float s_scal[32];  // [5]=reward [8..11]=logits [16..19]=s_log
                                // [20..23]=s_prob [24..27]=actionSel
  const int b = blockIdx.x;
  const int tid = threadIdx.x;
  const int lane = tid & 31, wave = tid >> 5;
  float* Ab   = ws + WS_A + (size_t)b * H * H;
  float* wrh  = ws + WS_RH;
  float* wrp  = ws + WS_RP;
  float* Ghpc = ws + WS_GHPC;
  float* Grp  = ws + WS_GRP;
  float* Grh  = ws + WS_GRH;
  float* Ge   = ws + WS_GE;
  unsigned* cnt = (unsigned*)(ws + WS_BAR);
  unsigned* gen = cnt + 1;
  const int nbase = b * 16;

  // decay = max(-sigmoid(-min(l,6)), -1/DT)
  const float l0 = lparam[0];
  const float decay = fmaxf(-1.0f / (1.0f + expf(fminf(l0, 6.0f))), -10.0f);
  const float fac = 1.0f + 0.1f * decay;

  // init: private copy of A (input must not be mutated), rh state
  {
    const float4* src = (const float4*)(A_init + (size_t)b * H * H);
    float4* dst = (float4*)Ab;
    for (int i = tid; i < H * H / 4; i += NTHR) dst[i] = src[i];
  }
  for (int i = tid; i < H; i += NTHR) {
    float r = rh_init[b * H + i];
    s_rh[i] = r;
    wrh[b * H + i] = r;
  }

  for (int trial = 0; trial < NTRI; ++trial) {
    for (int i = tid; i < H; i += NTHR) { s_rp[i] = 0.f; wrp[b * H + i] = 0.f; }
    if (tid < 12) s_scal[16 + tid] = 0.f;   // s_log, s_prob, actionSel
    if (tid == 0) s_scal[5] = 0.f;          // reward
    gbar(cnt, gen);

    for (int t = 0; t < TSTEPS; ++t) {
      // ---- G1: cooperative WMMA GEMMs on current state -------------------
      coop_gemm(wrp, W_h2h_hpc, Ghpc, nbase, s_gemm);
      coop_gemm(wrp, W_hh_pfc,  Grp,  nbase, s_gemm);
      coop_gemm(wrh, W_h2h_pfc, Grh,  nbase, s_gemm);
      gbar(cnt, gen);

      // ---- owner phase: stats of rh, A pass 1 ----------------------------
      float xa = s_rh[tid] + s_rh[tid + 256];
      float xq = s_rh[tid] * s_rh[tid] + s_rh[tid + 256] * s_rh[tid + 256];
      float m    = blk_sum(xa, s_red) * (1.f / 512.f);
      float invs = rsqrtf(blk_sum(xq, s_red) * (1.f / 512.f) - m * m + 1e-6f);
      for (int i = tid; i < H; i += NTHR) s_pre[i] = (s_rh[i] - m) * invs;
      __syncthreads();

      // A@rh and rowsum(A) in one L2-resident pass; wave-per-row, float4
      for (int r = wave * 64; r < wave * 64 + 64; ++r) {
        const float4* Ar = (const float4*)(Ab + r * H);
        float s1 = 0.f, s0 = 0.f;
#pragma unroll
        for (int k = 0; k < 4; ++k) {
          float4 a = Ar[lane + 32 * k];
          int j = 4 * (lane + 32 * k);
          s1 += a.x * s_rh[j] + a.y * s_rh[j + 1] + a.z * s_rh[j + 2] + a.w * s_rh[j + 3];
          s0 += a.x + a.y + a.z + a.w;
        }
#pragma unroll
        for (int o = 16; o; o >>= 1) { s1 += __shfl_xor(s1, o, 32); s0 += __shfl_xor(s0, o, 32); }
        if (lane == 0) { s_dotA[r] = s1; s_rowsum[r] = s0; }
      }
      // inp = [stim(12), reward, actionSel(4)]
      if (tid < 12) {
        float st = 0.f;
        if (t < 16) { int s = t >> 2; st = Xin[((b * 4 + s) * 3 + trial) * 12 + tid]; }
        s_inp[tid] = st;
      } else if (tid == 12) s_inp[12] = s_scal[5];
      else if (tid < 17)    s_inp[tid] = s_scal[24 + (tid - 13)];
      __syncthreads();

      // act_h -> rh_new, act_p -> rp_new
      for (int i = tid; i < H; i += NTHR) {
        float ah = (5.0f / 512.0f) * s_dotA[i];
        float ap = b_pfc[i];
        const float* wh = W_ih_hpc + i * NINP;
        const float* wp = W_ih_pfc + i * NINP;
#pragma unroll
        for (int j = 0; j < NINP; ++j) { float v = s_inp[j]; ah += wh[j] * v; ap += wp[j] * v; }
        ah += Ghpc[b * H + i];
        ah = fminf(fmaxf((ah - 0.5f) * 1.41421356237f, 0.f), 10.f);
        float rhn = 0.5f * s_rh[i] + 0.5f * ah;
        s_rhn[i] = rhn; wrh[b * H + i] = rhn;
        ap += Grp[b * H + i] + Grh[b * H + i];
        float rpn = 0.5f * s_rp[i] + 0.5f * fmaxf(ap, 0.f);
        s_rpn[i] = rpn; wrp[b * H + i] = rpn;
      }
      __syncthreads();
      float ya = s_rhn[tid] + s_rhn[tid + 256];
      float yq = s_rhn[tid] * s_rhn[tid] + s_rhn[tid + 256] * s_rhn[tid + 256];
      float m2    = blk_sum(ya, s_red) * (1.f / 512.f);
      float invs2 = rsqrtf(blk_sum(yq, s_red) * (1.f / 512.f) - m2 * m2 + 1e-6f);
      gbar(cnt, gen);

      // ---- cooperative GEMM for the error-gate path ----------------------
      coop_gemm(wrp, W_e, Ge, nbase, s_gemm);
      gbar(cnt, gen);

      // ---- owner phase: e, Hebbian A update, logits/behavior -------------
      float part = 0.f;
      for (int i = tid; i < H; i += NTHR)
        part += fmaxf(Ge[b * H + i] + b_e[i], 0.f) * W_pe[i];
      float dot = blk_sum(part, s_red);
      float ee = 0.01f / (1.0f + expf(-(dot + b_pe[0])));
      for (int i = tid; i < H; i += NTHR) {
        float post = (s_rhn[i] - m2) * invs2;
        float tmp  = (s_dotA[i] - m * s_rowsum[i]) * invs;   // A @ pre
        s_coef[i] = 0.1f * ee * (post - tmp);
      }
      __syncthreads();
      for (int r = wave * 64; r < wave * 64 + 64; ++r) {
        float cr = s_coef[r];
        float4* Ar = (float4*)(Ab + r * H);
#pragma unroll
        for (int k = 0; k < 4; ++k) {
          int idx = lane + 32 * k, j = 4 * idx;
          float4 a = Ar[idx];
          a.x = a.x * fac + cr * s_pre[j];
          a.y = a.y * fac + cr * s_pre[j + 1];
          a.z = a.z * fac + cr * s_pre[j + 2];
          a.w = a.w * fac + cr * s_pre[j + 3];
          Ar[idx] = a;
        }
      }
      if (wave < 4) {  // 4 logits, one wave each
        float p = 0.f;
        const float* wo = W_o + wave * H;
        for (int i = lane; i < H; i += 32) p += s_rpn[i] * wo[i];
#pragma unroll
        for (int o = 16; o; o >>= 1) p += __shfl_xor(p, o, 32);
        if (lane == 0) s_scal[8 + wave] = p + b_o[wave];
      }
      __syncthreads();
      if (tid == 0) {
        float rm = (t >= 12 && t < 20) ? 1.f : 0.f;
        float lg[4], mx = -1e30f;
        for (int j = 0; j < 4; ++j) { lg[j] = s_scal[8 + j]; mx = fmaxf(mx, lg[j]); }
        float pe[4], se = 0.f;
        for (int j = 0; j < 4; ++j) { pe[j] = expf(lg[j] - mx); se += pe[j]; }
        int am = 0; float best = -1e30f;
        for (int j = 0; j < 4; ++j) {
          s_scal[16 + j] += rm * lg[j];
          float np = s_scal[20 + j] + rm * pe[j] / se;
          s_scal[20 + j] = np;
          if (np > best) { best = np; am = j; }   // first-max tie break
        }
        if (t == 19) {
          float tgt = Yin[(b * 4 + am) * 3 + trial];
          float rew = (tgt > 0.9f) ? 1.f : -1.f;
          s_scal[5] = rew;
          for (int j = 0; j < 4; ++j) s_scal[24 + j] = (j == am) ? 1.f : 0.f;
        }
      }
      __syncthreads();
      for (int i = tid; i < H; i += NTHR) { s_rh[i] = s_rhn[i]; s_rp[i] = s_rpn[i]; }
      __syncthreads();
    }
    if (tid < 4) out[(trial * NBLK + b) * 4 + tid] = s_scal[16 + tid] * 0.125f;
    __syncthreads();
  }
}

extern "C" void kernel_launch(void* const* d_in, const int* in_sizes, int n_in,
                              void* d_out, int out_size, void* d_ws, size_t ws_size,
                              hipStream_t stream) {
  (void)in_sizes; (void)n_in; (void)out_size; (void)ws_size;
  const float* Xin      = (const float*)d_in[0];
  const float* Yin      = (const float*)d_in[1];
  const float* A_init   = (const float*)d_in[2];
  const float* rh_init  = (const float*)d_in[3];
  // d_in[4] = cc (does not influence the output)
  const float* W_ih_pfc = (const float*)d_in[5];
  const float* W_hh_pfc = (const float*)d_in[6];
  const float* b_pfc    = (const float*)d_in[7];
  const float* W_h2h_pfc= (const float*)d_in[8];
  const float* W_ih_hpc = (const float*)d_in[9];
  const float* W_h2h_hpc= (const float*)d_in[10];
  const float* lparam   = (const float*)d_in[11];
  const float* W_e      = (const float*)d_in[12];
  const float* b_e      = (const float*)d_in[13];
  const float* W_pe     = (const float*)d_in[14];
  const float* b_pe     = (const float*)d_in[15];
  const float* W_o      = (const float*)d_in[16];
  const float* b_o      = (const float*)d_in[17];
  float* ws = (float*)d_ws;
  unsigned* bar = (unsigned*)(ws + WS_BAR);
  wcst_init_bar<<<1, 1, 0, stream>>>(bar);
  wcst_persistent<<<NBLK, NTHR, 0, stream>>>(
      Xin, Yin, A_init, rh_init, W_ih_pfc, W_hh_pfc, b_pfc, W_h2h_pfc,
      W_ih_hpc, W_h2h_hpc, lparam, W_e, b_e, W_pe, b_pe, W_o, b_o,
      (float*)d_out, ws);
}